// AnchorBankCAA_12455405158619
// MI455X (gfx1250) — compile-verified
//
#include <hip/hip_runtime.h>

#define NUM_CLASSES 128
#define NUM_DOMAINS 6
#define FEAT_DIM    256
#define BATCH       524288
#define NSEG        (NUM_DOMAINS * NUM_CLASSES)   // 768
#define MOM         0.9f

typedef float v2f __attribute__((ext_vector_type(2)));
typedef float v8f __attribute__((ext_vector_type(8)));

// ---------------- Pass 1: segment histogram (LDS-privatized) ----------------
__global__ void k_hist(const int* __restrict__ y, const int* __restrict__ d,
                       int* __restrict__ seg_cnt) {
    __shared__ int lh[NSEG];
    for (int i = threadIdx.x; i < NSEG; i += blockDim.x) lh[i] = 0;
    __syncthreads();
    int stride = gridDim.x * blockDim.x;
    for (int i = blockIdx.x * blockDim.x + threadIdx.x; i < BATCH; i += stride) {
        int s = d[i] * NUM_CLASSES + y[i];
        atomicAdd(&lh[s], 1);
    }
    __syncthreads();
    for (int i = threadIdx.x; i < NSEG; i += blockDim.x)
        if (lh[i]) atomicAdd(&seg_cnt[i], lh[i]);
}

// ---------------- Pass 2: exclusive scan over 768 counts (one block) --------
__global__ void k_scan(const int* __restrict__ seg_cnt,
                       int* __restrict__ seg_off, int* __restrict__ cursor) {
    __shared__ int buf[NSEG];
    int t = threadIdx.x;
    if (t < NSEG) buf[t] = seg_cnt[t];
    __syncthreads();
    for (int off = 1; off < NSEG; off <<= 1) {   // Hillis-Steele inclusive scan
        int v = 0;
        if (t < NSEG && t >= off) v = buf[t - off];
        __syncthreads();
        if (t < NSEG) buf[t] += v;
        __syncthreads();
    }
    if (t < NSEG) {
        int excl = buf[t] - seg_cnt[t];
        seg_off[t] = excl;
        cursor[t]  = excl;
    }
}

// ---------------- Pass 3: scatter row ids into segment bins -----------------
__global__ void k_scatter(const int* __restrict__ y, const int* __restrict__ d,
                          int* __restrict__ cursor, int* __restrict__ order) {
    int i = blockIdx.x * blockDim.x + threadIdx.x;
    if (i >= BATCH) return;
    int s = d[i] * NUM_CLASSES + y[i];
    int pos = atomicAdd(&cursor[s], 1);
    order[pos] = i;
}

// ---- Pass 4: per-segment streaming reduce + anchor EMA + domain partials ---
__global__ void k_segreduce(const float* __restrict__ mu,
                            const float* __restrict__ anchors,
                            const int* __restrict__ seg_off,
                            const int* __restrict__ seg_cnt,
                            const int* __restrict__ order,
                            float* __restrict__ out_anchors,
                            float* __restrict__ d_sum,
                            float* __restrict__ d_sq) {
    int s   = blockIdx.x;          // segment = domain*128 + class
    int t   = threadIdx.x;         // feature dim
    int dom = s / NUM_CLASSES;
    int off = seg_off[s];
    int cnt = seg_cnt[s];
    __shared__ int rowbuf[256];
    float sum = 0.f, sq = 0.f;
    for (int base = 0; base < cnt; base += 256) {
        int n = min(256, cnt - base);
        if (t < n) rowbuf[t] = order[off + base + t];
        __syncthreads();
        for (int r = 0; r < n; ++r) {
            size_t row = (size_t)rowbuf[r];
            float v = mu[row * FEAT_DIM + t];   // coalesced across the block
            sum += v; sq += v * v;
        }
        __syncthreads();
    }
    float a    = anchors[(size_t)s * FEAT_DIM + t];
    float mean = sum / (float)max(cnt, 1);
    float na   = (cnt > 0) ? (MOM * a + (1.f - MOM) * mean) : a;
    out_anchors[(size_t)s * FEAT_DIM + t] = na;
    atomicAdd(&d_sum[dom * FEAT_DIM + t], sum);
    atomicAdd(&d_sq [dom * FEAT_DIM + t], sq);
}

// ---- Pass 5: per-domain mean/var EMA + all stats-align losses (1 block) ----
__global__ void k_domstats(const float* __restrict__ dm, const float* __restrict__ dv,
                           const int* __restrict__ seg_cnt,
                           const float* __restrict__ d_sum, const float* __restrict__ d_sq,
                           float* __restrict__ out_means, float* __restrict__ out_vars,
                           float* __restrict__ d_loss) {
    __shared__ float dcnt[NUM_DOMAINS];
    __shared__ float red[256];
    int t = threadIdx.x;
    if (t < NUM_DOMAINS) {
        int c = 0;
        for (int i = 0; i < NUM_CLASSES; ++i) c += seg_cnt[t * NUM_CLASSES + i];
        dcnt[t] = (float)c;
    }
    __syncthreads();
    float nm[NUM_DOMAINS], nv[NUM_DOMAINS];
    float tot_sum = 0.f, tot_sq = 0.f;
    for (int d = 0; d < NUM_DOMAINS; ++d) {
        float cnt = dcnt[d];
        float ds = d_sum[d * FEAT_DIM + t];
        float dq = d_sq [d * FEAT_DIM + t];
        tot_sum += ds; tot_sq += dq;
        float safe = fmaxf(cnt, 1.f);
        float bm = ds / safe;
        float bv = (dq - safe * bm * bm) / fmaxf(cnt - 1.f, 1.f);
        float m0 = dm[d * FEAT_DIM + t], v0 = dv[d * FEAT_DIM + t];
        bool g = cnt > 1.f;
        nm[d] = g ? MOM * m0 + (1.f - MOM) * bm : m0;
        nv[d] = g ? MOM * v0 + (1.f - MOM) * bv : v0;
        out_means[d * FEAT_DIM + t] = nm[d];
        out_vars [d * FEAT_DIM + t] = nv[d];
    }
    float gm = 0.f, gv = 0.f;
    for (int d = 0; d < NUM_DOMAINS; ++d) { gm += nm[d]; gv += nv[d]; }
    gm *= (1.f / NUM_DOMAINS); gv *= (1.f / NUM_DOMAINS);
    float lm = 0.f, lv = 0.f;
    for (int d = 0; d < NUM_DOMAINS; ++d) {
        float a = nm[d] - gm, b = nv[d] - gv;
        lm += a * a; lv += b * b;
    }
    float mu_mean = tot_sum / (float)BATCH;                 // sum over domains == global
    float mu_var  = tot_sq  / (float)BATCH - mu_mean * mu_mean;
    float a = mu_mean - gm, b = mu_var - gv;
    red[t] = (lm + lv) / (float)(NUM_DOMAINS * FEAT_DIM)
           + (a * a + b * b) / (float)FEAT_DIM;
    __syncthreads();
    for (int s2 = 128; s2 > 0; s2 >>= 1) {
        if (t < s2) red[t] += red[t + s2];
        __syncthreads();
    }
    if (t == 0) atomicAdd(d_loss, red[0]);
}

// ---- Pass 6: class means over domains, ||A_mean||^2, intra loss ------------
__global__ void k_amean(const float* __restrict__ new_anchors,
                        float* __restrict__ A, float* __restrict__ sqv,
                        float* __restrict__ d_loss) {
    int c = blockIdx.x, t = threadIdx.x;
    __shared__ float red[256];
    float v[NUM_DOMAINS];
    float am = 0.f;
    for (int d = 0; d < NUM_DOMAINS; ++d) {
        v[d] = new_anchors[((size_t)d * NUM_CLASSES + c) * FEAT_DIM + t];
        am += v[d];
    }
    am *= (1.f / NUM_DOMAINS);
    A[c * FEAT_DIM + t] = am;
    float dev = 0.f;
    for (int d = 0; d < NUM_DOMAINS; ++d) { float e = v[d] - am; dev += e * e; }
    red[t] = dev;
    __syncthreads();
    for (int s = 128; s > 0; s >>= 1) { if (t < s) red[t] += red[t + s]; __syncthreads(); }
    float devsum = red[0];
    __syncthreads();
    red[t] = am * am;
    __syncthreads();
    for (int s = 128; s > 0; s >>= 1) { if (t < s) red[t] += red[t + s]; __syncthreads(); }
    if (t == 0) {
        sqv[c] = red[0];
        atomicAdd(d_loss, devsum / (float)(NUM_DOMAINS * NUM_CLASSES * FEAT_DIM));
    }
}

// ---- Pass 7: WMMA f32 GEMM  G = A_mean @ A_mean^T  + hinge distance loss ---
// One wave32 per 16x16 output tile; V_WMMA_F32_16X16X4_F32, K stepped by 4.
__global__ void __launch_bounds__(32)
k_wmma_inter(const float* __restrict__ A, const float* __restrict__ sqv,
             float* __restrict__ d_loss) {
    int ti = blockIdx.x >> 3, tj = blockIdx.x & 7;
    int lane = threadIdx.x;          // 0..31, wave32
    int half = lane >> 4;            // K sub-group per ISA A/B 16x4 layout
    int lrow = lane & 15;
    const float* Ai = A + (size_t)(ti * 16 + lrow) * FEAT_DIM + half * 2;
    const float* Bj = A + (size_t)(tj * 16 + lrow) * FEAT_DIM + half * 2;
    v8f c = {0.f, 0.f, 0.f, 0.f, 0.f, 0.f, 0.f, 0.f};
    for (int kb = 0; kb < FEAT_DIM; kb += 4) {
        v2f a, b;
        a.x = Ai[kb]; a.y = Ai[kb + 1];     // lanes 0-15: K=kb,kb+1 ; 16-31: K=kb+2,kb+3
        b.x = Bj[kb]; b.y = Bj[kb + 1];
        c = __builtin_amdgcn_wmma_f32_16x16x4_f32(
                /*neg_a=*/false, a, /*neg_b=*/false, b,
                /*c_mod=*/(short)0, c, /*reuse_a=*/false, /*reuse_b=*/false);
    }
    float acc = 0.f;
    int   N   = tj * 16 + lrow;
    float sqN = sqv[N];
    for (int r = 0; r < 8; ++r) {           // C layout: M = r + 8*half, N = lane%16
        int M = ti * 16 + r + half * 8;
        if (M != N) {
            float d2   = sqv[M] + sqN - 2.f * c[r];
            float dist = sqrtf(fmaxf(d2, 1e-12f));
            acc += fmaxf(1.0f - dist, 0.f); // relu(MARGIN - dist)
        }
    }
    for (int off = 16; off > 0; off >>= 1) acc += __shfl_down(acc, off, 32);
    if (lane == 0)
        atomicAdd(d_loss, acc / (float)(NUM_CLASSES * (NUM_CLASSES - 1)));
}

// ---------------------------------------------------------------------------
extern "C" void kernel_launch(void* const* d_in, const int* in_sizes, int n_in,
                              void* d_out, int out_size, void* d_ws, size_t ws_size,
                              hipStream_t stream) {
    (void)in_sizes; (void)n_in; (void)out_size; (void)ws_size;
    const float* mu      = (const float*)d_in[0];
    const float* anchors = (const float*)d_in[1];
    const float* dmeans  = (const float*)d_in[2];
    const float* dvars   = (const float*)d_in[3];
    const int*   y       = (const int*)d_in[4];
    const int*   dt      = (const int*)d_in[5];

    float* out         = (float*)d_out;
    float* out_anchors = out;                                  // 6*128*256
    float* out_means   = out_anchors + (size_t)NSEG * FEAT_DIM;
    float* out_vars    = out_means + NUM_DOMAINS * FEAT_DIM;
    float* out_loss    = out_vars + NUM_DOMAINS * FEAT_DIM;    // scalar

    char* ws = (char*)d_ws;
    int*   order   = (int*)ws;   ws += (size_t)BATCH * 4;           // 2 MB
    int*   seg_cnt = (int*)ws;   ws += NSEG * 4;
    int*   seg_off = (int*)ws;   ws += NSEG * 4;
    int*   cursor  = (int*)ws;   ws += NSEG * 4;
    float* d_sum   = (float*)ws; ws += NUM_DOMAINS * FEAT_DIM * 4;
    float* d_sq    = (float*)ws; ws += NUM_DOMAINS * FEAT_DIM * 4;
    float* Amean   = (float*)ws; ws += NUM_CLASSES * FEAT_DIM * 4;  // 128 KB
    float* sqv     = (float*)ws; ws += NUM_CLASSES * 4;

    hipMemsetAsync(seg_cnt, 0, NSEG * 4, stream);
    hipMemsetAsync(d_sum, 0, NUM_DOMAINS * FEAT_DIM * 4 * 2, stream); // d_sum+d_sq
    hipMemsetAsync(out_loss, 0, 4, stream);

    k_hist     <<<256, 256, 0, stream>>>(y, dt, seg_cnt);
    k_scan     <<<1, 1024, 0, stream>>>(seg_cnt, seg_off, cursor);
    k_scatter  <<<BATCH / 256, 256, 0, stream>>>(y, dt, cursor, order);
    k_segreduce<<<NSEG, 256, 0, stream>>>(mu, anchors, seg_off, seg_cnt, order,
                                          out_anchors, d_sum, d_sq);
    k_domstats <<<1, 256, 0, stream>>>(dmeans, dvars, seg_cnt, d_sum, d_sq,
                                       out_means, out_vars, out_loss);
    k_amean    <<<NUM_CLASSES, 256, 0, stream>>>(out_anchors, Amean, sqv, out_loss);
    k_wmma_inter<<<64, 32, 0, stream>>>(Amean, sqv, out_loss);
}